// HeterogeneousGraphAttentionNetwork_69561290326872
// MI455X (gfx1250) — compile-verified
//
#include <hip/hip_runtime.h>

// ---------------- types ----------------
typedef __attribute__((ext_vector_type(2))) float v2f;
typedef __attribute__((ext_vector_type(8))) float v8f;

#define N_OP  50000
#define N_MA  1000
#define N_JOB 5000
#define HID   128

// ---------------- utility ----------------
__global__ void fill_kernel(float* __restrict__ p, float v, int n) {
    int i = blockIdx.x * blockDim.x + threadIdx.x;
    if (i < n) p[i] = v;
}

__device__ __forceinline__ void atomicMaxFloat(float* addr, float val) {
    // monotone int/uint trick; addr initialized to -inf
    if (val >= 0.0f) atomicMax((int*)addr, __float_as_int(val));
    else             atomicMin((unsigned int*)addr, __float_as_uint(val));
}

// ---------------- WMMA f32 GEMM:  C[M x 128] = A[M x K] @ W[K x 128] (+ bias) ----------------
// block: 256 threads = 8 wave32; each wave computes a 16x128 strip.
// W staged in LDS in fragment-native layout: for k-group g (4 K-values) and n-tile,
// the 64 floats are stored exactly as the 32 lanes' v2f B-fragments:
//   Wlds[(g*8+n)*64 + lane*2 + pair], lane = (kk>>1)*16 + (col&15), pair = kk&1
// -> each B fragment is ONE ds_load_b64 (dwords 0..63 across lanes: all 64 banks).
// All 8 B fragments of a k-group are fetched into registers BEFORE the WMMA chain
// so the ds loads issue as a batch and drain with staged s_wait_dscnt.
// A frag (16x4 f32): lanes 0-15 hold M rows, v.x=K0,v.y=K1 ; lanes 16-31: v.x=K2,v.y=K3
// C/D (16x16 f32): acc[v] = C[M = v + 8*(lane>=16)][N = lane&15]
extern "C" __global__ __launch_bounds__(256)
void gemm_wmma_f32(const float* __restrict__ A, const float* __restrict__ W,
                   const float* __restrict__ bias, float* __restrict__ C,
                   int M, int K, int has_bias)
{
    extern __shared__ float Wlds[];                 // K*128 floats (permuted)
    {
        const int total4 = (K * 128) >> 2;
        const float4* W4 = (const float4*)W;
        for (int i = threadIdx.x; i < total4; i += 256) {
            const float4 wv = W4[i];
            const int idx = i << 2;
            const int k   = idx >> 7;               // row of W
            const int col = idx & 127;              // 4-aligned column
            const int g = k >> 2, kk = k & 3;
            const int n = col >> 4, l16 = col & 15;
            const int base = (g * 8 + n) * 64 + ((kk >> 1) * 16) * 2 + (kk & 1);
            Wlds[base + (l16 + 0) * 2] = wv.x;
            Wlds[base + (l16 + 1) * 2] = wv.y;
            Wlds[base + (l16 + 2) * 2] = wv.z;
            Wlds[base + (l16 + 3) * 2] = wv.w;
        }
    }
    __syncthreads();

    const int wave = threadIdx.x >> 5;
    const int lane = threadIdx.x & 31;
    const int half = lane >> 4;                      // 0 | 1
    const int l16  = lane & 15;
    const int rowbase = blockIdx.x * 128 + wave * 16;

    v8f acc[8] = {};

    const int   arow = rowbase + l16;
    const float msk  = (arow < M) ? 1.0f : 0.0f;     // branch-free OOB handling
    const float* Arow = A + (size_t)((arow < M) ? arow : (M - 1)) * K;

    const int NG = K >> 2;
    for (int g = 0; g < NG; ++g) {
        v2f a = *(const v2f*)(Arow + (g << 2) + half * 2);   // one global_load_b64
        a.x *= msk; a.y *= msk;
        const float* wb = Wlds + g * 512 + lane * 2;
        v2f bf[8];
#pragma unroll
        for (int n = 0; n < 8; ++n) bf[n] = *(const v2f*)(wb + n * 64);  // batched ds_load_b64
#pragma unroll
        for (int n = 0; n < 8; ++n) {
            acc[n] = __builtin_amdgcn_wmma_f32_16x16x4_f32(
                false, a, false, bf[n], (short)0, acc[n], false, false);
        }
    }

    const int rowoff = half * 8;
#pragma unroll
    for (int n = 0; n < 8; ++n) {
        const int col = n * 16 + l16;
        const float bv = has_bias ? bias[col] : 0.0f;
#pragma unroll
        for (int v = 0; v < 8; ++v) {
            const int row = rowbase + rowoff + v;
            if (row < M) C[(size_t)row * 128 + col] = acc[n][v] + bv;
        }
    }
}

// ---------------- fold edge-attention:  U[i,t] = We_proj[t] @ (Wge[i,t] . att_e heads), cvec = be_proj[t] . V ----------------
extern "C" __global__ __launch_bounds__(256)
void fold_edge_att_kernel(const float* __restrict__ We_proj,  // [5][32][128]
                          const float* __restrict__ be_proj,  // [5][128]
                          const float* __restrict__ Wge,      // [3][5][128][128]
                          const float* __restrict__ att_e,    // [3][5][128]
                          float* __restrict__ U,              // [3][5][32*8] packed [k][h], stride H
                          float* __restrict__ cvec)           // [3][5][8]
{
    const int it = blockIdx.x;           // 0..14
    const int layer = it / 5, t = it % 5;
    const int H = (layer < 2) ? 8 : 1;
    const int Cc = 128 / H;
    __shared__ float V[128 * 8];

    const float* Wge_p = Wge + (size_t)(layer * 5 + t) * 128 * 128;
    const float* ae_p  = att_e + (size_t)(layer * 5 + t) * 128;
    for (int idx = threadIdx.x; idx < 128 * H; idx += 256) {
        const int d = idx / H, h = idx % H;
        float s = 0.0f;
        for (int c = 0; c < Cc; ++c) s += Wge_p[d * 128 + h * Cc + c] * ae_p[h * Cc + c];
        V[d * H + h] = s;
    }
    __syncthreads();

    const float* Wep = We_proj + (size_t)t * 32 * 128;
    float* Up = U + (size_t)(layer * 5 + t) * 256;
    for (int idx = threadIdx.x; idx < 32 * H; idx += 256) {
        const int k = idx / H, h = idx % H;
        float s = 0.0f;
        for (int d = 0; d < 128; ++d) s += Wep[k * 128 + d] * V[d * H + h];
        Up[k * H + h] = s;
    }
    const float* bep = be_proj + (size_t)t * 128;
    float* cp = cvec + (size_t)(layer * 5 + t) * 8;
    for (int h = threadIdx.x; h < H; h += 256) {
        float s = 0.0f;
        for (int d = 0; d < 128; ++d) s += bep[d] * V[d * H + h];
        cp[h] = s;
    }
}

// ---------------- per-node head dots:  out[n,h] = sum_c proj[n, h*C+c] * att[h*C+c] ----------------
extern "C" __global__ void head_dot_kernel(const float* __restrict__ proj,
                                           const float* __restrict__ att,
                                           float* __restrict__ out, int N, int H, int C)
{
    const int idx = blockIdx.x * blockDim.x + threadIdx.x;
    if (idx >= N * H) return;
    const int n = idx / H, h = idx % H;
    const float* p = proj + (size_t)n * 128 + h * C;
    const float* a = att + h * C;
    float s = 0.0f;
    for (int c = 0; c < C; ++c) s += p[c] * a[c];
    out[idx] = s;
}

// ---------------- edge pass A: logits + segment max ----------------
extern "C" __global__ void edge_logits_max_kernel(const int* __restrict__ ei,
                                                  const float* __restrict__ ea,   // [E][32] raw
                                                  const float* __restrict__ U,    // [32][H]
                                                  const float* __restrict__ cvec, // [H]
                                                  const float* __restrict__ a_s,
                                                  const float* __restrict__ a_d,
                                                  float* __restrict__ logits,
                                                  float* __restrict__ segmax,
                                                  int E, int H)
{
    const int e = blockIdx.x * blockDim.x + threadIdx.x;
    if (e >= E) return;
    const int src = ei[e], dst = ei[E + e];
    // keep the 32-float edge-attr row register-resident across heads
    const float4* eap4 = (const float4*)(ea + (size_t)e * 32);
    float4 ev[8];
#pragma unroll
    for (int q = 0; q < 8; ++q) ev[q] = eap4[q];

    for (int h = 0; h < H; ++h) {
        float ae = cvec[h];
#pragma unroll
        for (int q = 0; q < 8; ++q) {
            const int k = q * 4;
            ae += ev[q].x * U[(k + 0) * H + h];
            ae += ev[q].y * U[(k + 1) * H + h];
            ae += ev[q].z * U[(k + 2) * H + h];
            ae += ev[q].w * U[(k + 3) * H + h];
        }
        float x = a_s[src * H + h] + a_d[dst * H + h] + ae;
        float lg = (x > 0.0f) ? x : 0.2f * x;          // leaky_relu(0.2)
        logits[(size_t)e * H + h] = lg;
        atomicMaxFloat(&segmax[dst * H + h], lg);
    }
}

// ---------------- edge pass B: exp(logit - max), segment sum; logits buffer becomes exp ----------------
extern "C" __global__ void edge_exp_sum_kernel(const int* __restrict__ ei,
                                               float* __restrict__ exl,
                                               const float* __restrict__ segmax,
                                               float* __restrict__ segsum,
                                               int E, int H)
{
    const int idx = blockIdx.x * blockDim.x + threadIdx.x;
    if (idx >= E * H) return;
    const int e = idx / H, h = idx % H;
    const int dst = ei[E + e];
    const float ex = __expf(exl[idx] - segmax[dst * H + h]);
    exl[idx] = ex;
    atomicAdd(&segsum[dst * H + h], ex);
}

// ---------------- edge pass C: alpha-weighted gather/scatter of messages ----------------
extern "C" __global__ void edge_scatter_kernel(const int* __restrict__ ei,
                                               const float* __restrict__ exl,
                                               const float* __restrict__ segsum,
                                               const float* __restrict__ ps,   // proj of src type [Ns][128]
                                               float* __restrict__ agg,        // [Nd][128]
                                               int E, int H, int logC)
{
    const int idx = blockIdx.x * blockDim.x + threadIdx.x;
    const int e = idx >> 2;
    if (e >= E) return;
    const int q = idx & 3;
    const int src = ei[e], dst = ei[E + e];
    const float4* sp4 = (const float4*)(ps + (size_t)src * 128 + q * 32);
    float* dp = agg + (size_t)dst * 128 + q * 32;
#pragma unroll
    for (int c4 = 0; c4 < 8; ++c4) {
        const float4 v = sp4[c4];
        const int col0 = q * 32 + c4 * 4;
        const int h = col0 >> logC;                    // C>=16, 4-aligned cols share a head
        const float alpha = exl[(size_t)e * H + h] / (segsum[dst * H + h] + 1e-16f);
        atomicAdd(dp + c4 * 4 + 0, v.x * alpha);
        atomicAdd(dp + c4 * 4 + 1, v.y * alpha);
        atomicAdd(dp + c4 * 4 + 2, v.z * alpha);
        atomicAdd(dp + c4 * 4 + 3, v.w * alpha);
    }
}

// ---------------- BatchNorm stats: per-column sum & sumsq (bg bias cancels in BN, skipped) ----------------
extern "C" __global__ __launch_bounds__(256)
void bn_stats_kernel(const float* __restrict__ h, float* __restrict__ stats, int N)
{
    const int col = blockIdx.x;                      // 128 blocks
    float s = 0.0f, s2 = 0.0f;
    for (int n = threadIdx.x; n < N; n += 256) {
        const float v = h[(size_t)n * 128 + col];
        s += v; s2 += v * v;
    }
    __shared__ float sh[256], sh2[256];
    sh[threadIdx.x] = s; sh2[threadIdx.x] = s2;
    __syncthreads();
    for (int off = 128; off > 0; off >>= 1) {
        if (threadIdx.x < off) { sh[threadIdx.x] += sh[threadIdx.x + off]; sh2[threadIdx.x] += sh2[threadIdx.x + off]; }
        __syncthreads();
    }
    if (threadIdx.x == 0) {
        const float mu = sh[0] / (float)N;
        stats[col]       = mu;
        stats[128 + col] = sh2[0] / (float)N - mu * mu;   // biased variance
    }
}

// ---------------- BN apply + optional ReLU + residual ----------------
extern "C" __global__ void bn_apply_kernel(const float* __restrict__ agg,
                                           const float* __restrict__ stats,
                                           const float* __restrict__ gamma,
                                           const float* __restrict__ beta,
                                           const float* __restrict__ xold,
                                           float* __restrict__ xnew,
                                           int N, int do_relu)
{
    const int idx = blockIdx.x * blockDim.x + threadIdx.x;
    if (idx >= N * 128) return;
    const int col = idx & 127;
    const float mu  = stats[col];
    const float var = stats[128 + col];
    float v = (agg[idx] - mu) * rsqrtf(var + 1e-5f) * gamma[col] + beta[col];
    if (do_relu && v < 0.0f) v = 0.0f;
    xnew[idx] = v + xold[idx];
}

// =================================================================================
extern "C" void kernel_launch(void* const* d_in, const int* in_sizes, int n_in,
                              void* d_out, int out_size, void* d_ws, size_t ws_size,
                              hipStream_t stream)
{
    const float* x_op   = (const float*)d_in[0];
    const float* x_ma   = (const float*)d_in[1];
    const float* x_job  = (const float*)d_in[2];
    const int*   ei[5]  = {(const int*)d_in[3], (const int*)d_in[5], (const int*)d_in[7],
                           (const int*)d_in[9], (const int*)d_in[11]};
    const float* ea[5]  = {(const float*)d_in[4], (const float*)d_in[6], (const float*)d_in[8],
                           (const float*)d_in[10], (const float*)d_in[12]};
    const float* Wn_op  = (const float*)d_in[13];
    const float* bn_op  = (const float*)d_in[14];
    const float* Wn_ma  = (const float*)d_in[15];
    const float* bn_ma  = (const float*)d_in[16];
    const float* Wn_job = (const float*)d_in[17];
    const float* bn_job = (const float*)d_in[18];
    const float* We_proj= (const float*)d_in[19];
    const float* be_proj= (const float*)d_in[20];
    const float* Wg     = (const float*)d_in[21];
    const float* att_s  = (const float*)d_in[22];
    const float* att_d  = (const float*)d_in[23];
    const float* Wge    = (const float*)d_in[24];
    const float* att_e  = (const float*)d_in[25];
    /* d_in[26] = bg: provably cancels under BatchNorm (uniform per-column shift) */
    const float* gamma  = (const float*)d_in[27];
    const float* beta   = (const float*)d_in[28];
    const float* Wo     = (const float*)d_in[29];
    const float* bo     = (const float*)d_in[30];

    const int Ns[3]   = {N_OP, N_MA, N_JOB};
    const int Esz[5]  = {150000, 100000, 150000, 100000, 100000};
    const int stv[5]  = {0, 0, 1, 2, 0};   // src node type
    const int dtv[5]  = {0, 1, 0, 0, 2};   // dst node type

    // ---- workspace layout (floats) ----
    float* w = (float*)d_ws;
    size_t off = 0;
    auto alloc = [&](size_t n) { float* p = w + off; off += n; return p; };
    float* xA[3]; float* xB[3]; float* agg[3];
    for (int nt = 0; nt < 3; ++nt) xA[nt]  = alloc((size_t)Ns[nt] * HID);
    for (int nt = 0; nt < 3; ++nt) xB[nt]  = alloc((size_t)Ns[nt] * HID);
    for (int nt = 0; nt < 3; ++nt) agg[nt] = alloc((size_t)Ns[nt] * HID);
    float* projS  = alloc((size_t)N_OP * HID);
    float* projD  = alloc((size_t)N_OP * HID);
    float* a_sbuf = alloc((size_t)N_OP * 8);
    float* a_dbuf = alloc((size_t)N_OP * 8);
    float* logits = alloc((size_t)150000 * 8);
    float* segmax = alloc((size_t)N_OP * 8);
    float* segsum = alloc((size_t)N_OP * 8);
    float* stats  = alloc(256);
    float* Ubuf   = alloc(15 * 256);
    float* cbuf   = alloc(15 * 8);

    auto gemm = [&](const float* A, const float* Wm, const float* bias, float* Cout, int M, int K) {
        gemm_wmma_f32<<<(M + 127) / 128, 256, (size_t)K * 128 * sizeof(float), stream>>>(
            A, Wm, bias, Cout, M, K, bias != nullptr);
    };
    auto fill = [&](float* p, float v, int n) {
        fill_kernel<<<(n + 255) / 256, 256, 0, stream>>>(p, v, n);
    };

    // Phase 0: fold edge-attr attention vectors (tiny)
    fold_edge_att_kernel<<<15, 256, 0, stream>>>(We_proj, be_proj, Wge, att_e, Ubuf, cbuf);

    // Phase 1: node input projections -> HID
    gemm(x_op,  Wn_op,  bn_op,  xA[0], N_OP,  64);
    gemm(x_ma,  Wn_ma,  bn_ma,  xA[1], N_MA,  32);
    gemm(x_job, Wn_job, bn_job, xA[2], N_JOB, 16);

    float* xcur[3] = {xA[0], xA[1], xA[2]};
    float* xnxt[3] = {xB[0], xB[1], xB[2]};
    const float NEG_INF = -__builtin_inff();

    for (int i = 0; i < 3; ++i) {
        const int H = (i < 2) ? 8 : 1;
        const int C = HID / H;
        const int logC = (i < 2) ? 4 : 7;

        for (int nt = 0; nt < 3; ++nt) fill(agg[nt], 0.0f, Ns[nt] * HID);

        for (int t = 0; t < 5; ++t) {
            const int st = stv[t], dt = dtv[t];
            const int nS = Ns[st], nD = Ns[dt];
            const int E = Esz[t];
            const float* Wg_it = Wg + (size_t)(i * 5 + t) * HID * HID;

            gemm(xcur[st], Wg_it, nullptr, projS, nS, HID);
            float* pD = projS;
            if (st != dt) { gemm(xcur[dt], Wg_it, nullptr, projD, nD, HID); pD = projD; }

            head_dot_kernel<<<(nS * H + 255) / 256, 256, 0, stream>>>(
                projS, att_s + (size_t)(i * 5 + t) * HID, a_sbuf, nS, H, C);
            head_dot_kernel<<<(nD * H + 255) / 256, 256, 0, stream>>>(
                pD, att_d + (size_t)(i * 5 + t) * HID, a_dbuf, nD, H, C);

            fill(segmax, NEG_INF, nD * H);
            fill(segsum, 0.0f,    nD * H);

            edge_logits_max_kernel<<<(E + 255) / 256, 256, 0, stream>>>(
                ei[t], ea[t], Ubuf + (size_t)(i * 5 + t) * 256, cbuf + (size_t)(i * 5 + t) * 8,
                a_sbuf, a_dbuf, logits, segmax, E, H);
            edge_exp_sum_kernel<<<(E * H + 255) / 256, 256, 0, stream>>>(
                ei[t], logits, segmax, segsum, E, H);
            edge_scatter_kernel<<<(E * 4 + 255) / 256, 256, 0, stream>>>(
                ei[t], logits, segsum, projS, agg[dt], E, H, logC);
        }

        for (int nt = 0; nt < 3; ++nt) {
            bn_stats_kernel<<<128, 256, 0, stream>>>(agg[nt], stats, Ns[nt]);
            bn_apply_kernel<<<(Ns[nt] * HID + 255) / 256, 256, 0, stream>>>(
                agg[nt], stats,
                gamma + (size_t)(i * 3 + nt) * HID, beta + (size_t)(i * 3 + nt) * HID,
                xcur[nt], xnxt[nt], Ns[nt], (i < 2) ? 1 : 0);
        }
        for (int nt = 0; nt < 3; ++nt) { float* tmp = xcur[nt]; xcur[nt] = xnxt[nt]; xnxt[nt] = tmp; }
    }

    // Output projection: out = x_op @ Wo + bo
    gemm(xcur[0], Wo, bo, (float*)d_out, N_OP, HID);
}